// DynamicConv_1357209666397
// MI455X (gfx1250) — compile-verified
//
#include <hip/hip_runtime.h>
#include <hip/hip_bf16.h>

typedef float v2f __attribute__((ext_vector_type(2)));
typedef float v8f __attribute__((ext_vector_type(8)));

#define B_      32
#define CIN_    128
#define COUT_   128
#define NK_     8
#define H_      64
#define W_      64
#define HW_     4096
#define HIDDEN_ 32
#define TEMP_   30.0f

// ---------------- workspace layout (floats) ----------------
// pooled : [B_*CIN_]            @ 0        (4096)
// pi     : [B_*NK_]             @ 4096     (256)
// bmix   : [B_*COUT_]           @ 4352     (4096)
// wmix   : [B_][9][16384]       @ 8448     (4,718,592)
//          per (b,uv) tile is K-interleaved: element (ci,co) at
//          (ci>>2)*512 + co*4 + (ci&3)  -> A fragments are contiguous b64
#define WS_POOLED 0
#define WS_PI     4096
#define WS_BMIX   4352
#define WS_WMIX   8448

// ---------------- 1) global average pool: one block per (b,c) ----------------
__global__ void dyn_pool_kernel(const float* __restrict__ x, float* __restrict__ pooled) {
    __shared__ float red[256];
    int bc = blockIdx.x;                       // b*CIN_ + c
    const float* px = x + (size_t)bc * HW_;
    float s = 0.f;
    for (int i = threadIdx.x; i < HW_; i += 256) s += px[i];
    red[threadIdx.x] = s;
    __syncthreads();
    for (int off = 128; off > 0; off >>= 1) {
        if ((int)threadIdx.x < off) red[threadIdx.x] += red[threadIdx.x + off];
        __syncthreads();
    }
    if (threadIdx.x == 0) pooled[bc] = red[0] * (1.0f / HW_);
}

// ---------------- 2) attention MLP + softmax (single block) ----------------
__global__ void dyn_attn_kernel(const float* __restrict__ pooled,
                                const float* __restrict__ fc1w, const float* __restrict__ fc1b,
                                const float* __restrict__ fc2w, const float* __restrict__ fc2b,
                                float* __restrict__ pi) {
    __shared__ float sp[B_ * CIN_];
    __shared__ float sh[B_ * HIDDEN_];
    __shared__ float sl[B_ * NK_];
    int tid = threadIdx.x;
    for (int i = tid; i < B_ * CIN_; i += 256) sp[i] = pooled[i];
    __syncthreads();
    for (int i = tid; i < B_ * HIDDEN_; i += 256) {
        int b = i >> 5, j = i & 31;
        float acc = fc1b[j];
        for (int c = 0; c < CIN_; ++c) acc += sp[b * CIN_ + c] * fc1w[j * CIN_ + c];
        sh[i] = fmaxf(acc, 0.f);                               // ReLU
    }
    __syncthreads();
    {   // 256 logits: one per thread
        int b = tid >> 3, k = tid & 7;
        float acc = fc2b[k];
        for (int j = 0; j < HIDDEN_; ++j) acc += sh[b * HIDDEN_ + j] * fc2w[k * HIDDEN_ + j];
        sl[tid] = acc;
    }
    __syncthreads();
    if (tid < B_) {
        float m = -1e30f;
        for (int k = 0; k < NK_; ++k) m = fmaxf(m, sl[tid * NK_ + k]);
        float e[NK_], s = 0.f;
        for (int k = 0; k < NK_; ++k) { e[k] = __expf((sl[tid * NK_ + k] - m) * (1.0f / TEMP_)); s += e[k]; }
        float inv = 1.0f / s;
        for (int k = 0; k < NK_; ++k) pi[tid * NK_ + k] = e[k] * inv;
    }
}

// ---------------- 3) mix expert weights into K-interleaved tiles ----------------
// out flat (per b,uv tile): addr = ii*512 + co*4 + r  with ci = ii*4 + r
__global__ void dyn_wmix_kernel(const float* __restrict__ weight, const float* __restrict__ pi,
                                float* __restrict__ wmix) {
    int idx = blockIdx.x * 256 + threadIdx.x;   // < B_*9*16384
    int r    = idx & 3;
    int co   = (idx >> 2) & 127;
    int ii   = (idx >> 9) & 31;
    int rest = idx >> 14;
    int uv   = rest % 9;
    int b    = rest / 9;
    int ci   = ii * 4 + r;
    float s = 0.f;
#pragma unroll
    for (int k = 0; k < NK_; ++k)
        s += pi[b * NK_ + k] * weight[(((size_t)k * COUT_ + co) * CIN_ + ci) * 9 + uv];
    wmix[idx] = s;
}

__global__ void dyn_bmix_kernel(const float* __restrict__ bias, const float* __restrict__ pi,
                                float* __restrict__ bmix) {
    int idx = blockIdx.x * 256 + threadIdx.x;   // < B_*COUT_
    int b = idx >> 7, co = idx & 127;
    float s = 0.f;
#pragma unroll
    for (int k = 0; k < NK_; ++k) s += pi[b * NK_ + k] * bias[k * COUT_ + co];
    bmix[idx] = s;
}

// ---------------- 4) conv as 9 shifted GEMMs via V_WMMA_F32_16X16X4_F32 ----------------
// grid: B_ * (HW_/128) blocks of 256 threads (8 waves).
// Each block: sample b, 128-pixel N-tile. Wave w owns M rows [16w,16w+16) x 128 cols.
__global__ __launch_bounds__(256, 1)
void dyn_conv_kernel(const float* __restrict__ x, const float* __restrict__ wmix,
                     const float* __restrict__ bmix, float* __restrict__ out) {
    // K-interleaved tiles: element (k, col) at (k>>2)*512 + col*4 + (k&3)
    __shared__ float As[CIN_ * COUT_];   // 64 KiB (cols = co)
    __shared__ float Bs[CIN_ * 128];     // 64 KiB (cols = pixel)

    int b     = blockIdx.x >> 5;
    int nTile = (blockIdx.x & 31) * 128;
    int tid   = threadIdx.x;
    int wave  = tid >> 5;
    int lane  = tid & 31;
    int half  = lane >> 4;               // wave32: lanes 0-15 take K {0,1}, 16-31 take K {2,3}
    int l16   = lane & 15;
    int mBase = wave * 16;

    v8f acc[8];
#pragma unroll
    for (int j = 0; j < 8; ++j)
#pragma unroll
        for (int r = 0; r < 8; ++r) acc[j][r] = 0.0f;

    const float* xb = x + (size_t)b * CIN_ * HW_;
    const float* wbase = wmix + (size_t)b * 9 * (CIN_ * COUT_);

    const int aFragBase = (mBase + l16) * 4 + 2 * half;   // + kk4*512
    const int bFragBase = l16 * 4 + 2 * half;             // + kk4*512 + j*64

    for (int uv = 0; uv < 9; ++uv) {
        int du = uv / 3 - 1, dv = uv % 3 - 1;
        const float* wt = wbase + uv * (CIN_ * COUT_);

        __syncthreads();                 // previous iteration readers done
        // stage A-tile: contiguous copy (wmix already K-interleaved)
#pragma unroll 4
        for (int i = 0; i < 64; ++i) {
            int flat = tid + i * 256;
            As[flat] = wt[flat];
        }
        // stage B-tile: shifted image with zero padding, scatter to interleaved layout
#pragma unroll 4
        for (int i = 0; i < 64; ++i) {
            int flat = tid + i * 256;
            int ci = flat >> 7, p = flat & 127;
            int pix = nTile + p;
            int hh = (pix >> 6) + du;
            int ww = (pix & 63) + dv;
            float v = 0.f;
            if ((unsigned)hh < (unsigned)H_ && (unsigned)ww < (unsigned)W_)
                v = xb[ci * HW_ + hh * W_ + ww];
            Bs[(ci >> 2) * 512 + p * 4 + (ci & 3)] = v;
        }
        // prefetch next tap's A-tile into L2 while we compute (global_prefetch_b8)
        if (uv < 8) __builtin_prefetch(wt + CIN_ * COUT_ + tid * 64, 0, 0);
        __syncthreads();

        // K = 128: one live A fragment per K-step, 8 independent WMMAs
#pragma unroll 4
        for (int kk4 = 0; kk4 < 32; ++kk4) {
            v2f af = *(const v2f*)&As[kk4 * 512 + aFragBase];
            const float* bp = &Bs[kk4 * 512 + bFragBase];
#pragma unroll
            for (int j = 0; j < 8; ++j) {
                v2f bf = *(const v2f*)(bp + j * 64);
                acc[j] = __builtin_amdgcn_wmma_f32_16x16x4_f32(
                    false, af, false, bf, (short)0, acc[j], false, false);
            }
        }
    }

    // writeback: C/D layout — VGPR r holds M = r + 8*half, N = l16
#pragma unroll
    for (int j = 0; j < 8; ++j) {
#pragma unroll
        for (int r = 0; r < 8; ++r) {
            int co  = mBase + 8 * half + r;
            int pix = nTile + j * 16 + l16;
            out[((size_t)b * COUT_ + co) * HW_ + pix] = acc[j][r] + bmix[b * COUT_ + co];
        }
    }
}

// ---------------- launcher ----------------
extern "C" void kernel_launch(void* const* d_in, const int* in_sizes, int n_in,
                              void* d_out, int out_size, void* d_ws, size_t ws_size,
                              hipStream_t stream) {
    const float* x     = (const float*)d_in[0];
    const float* weight= (const float*)d_in[1];
    const float* bias  = (const float*)d_in[2];
    const float* fc1w  = (const float*)d_in[3];
    const float* fc1b  = (const float*)d_in[4];
    const float* fc2w  = (const float*)d_in[5];
    const float* fc2b  = (const float*)d_in[6];
    float* out = (float*)d_out;

    float* ws     = (float*)d_ws;
    float* pooled = ws + WS_POOLED;
    float* pi     = ws + WS_PI;
    float* bmix   = ws + WS_BMIX;
    float* wmixp  = ws + WS_WMIX;

    dyn_pool_kernel<<<B_ * CIN_, 256, 0, stream>>>(x, pooled);
    dyn_attn_kernel<<<1, 256, 0, stream>>>(pooled, fc1w, fc1b, fc2w, fc2b, pi);
    dyn_wmix_kernel<<<(B_ * 9 * CIN_ * COUT_) / 256, 256, 0, stream>>>(weight, pi, wmixp);
    dyn_bmix_kernel<<<(B_ * COUT_) / 256, 256, 0, stream>>>(bias, pi, bmix);
    dyn_conv_kernel<<<B_ * (HW_ / 128), 256, 0, stream>>>(x, wmixp, bmix, out);
}